// CausalSelfAttention_1949915152515
// MI455X (gfx1250) — compile-verified
//
#include <hip/hip_runtime.h>
#include <hip/hip_bf16.h>
#include <math.h>

// CausalSelfAttention (per-channel outer-product attention) for MI455X / gfx1250.
// Pipeline: [qkv f32-WMMA GEMMs] -> [heads-axis RoPE] -> [per-channel causal
// softmax rows on VALU, wave32 shuffle reductions] -> [output f32-WMMA GEMM].
// All tensors ~1MB each: L2-resident; latency-bound, so keep full f32 via
// V_WMMA_F32_16X16X4_F32 rather than downcasting.

#define TT 512
#define DD 512

typedef __attribute__((ext_vector_type(2))) float v2f;
typedef __attribute__((ext_vector_type(8))) float v8f;

// ---------------- Phase 1: q/k/v = W (DxD) x x^T (DxT) -> [D, T] ----------------
// C[o,t] = sum_d W[o,d] * x[t,d].  One wave computes one 16x16 tile of C.
__global__ void qkv_gemm_f32(const float* __restrict__ x,
                             const float* __restrict__ Wq,
                             const float* __restrict__ Wk,
                             const float* __restrict__ Wv,
                             float* __restrict__ q,
                             float* __restrict__ k,
                             float* __restrict__ v) {
    const float* W = (blockIdx.z == 0) ? Wq : (blockIdx.z == 1) ? Wk : Wv;
    float* dst     = (blockIdx.z == 0) ? q  : (blockIdx.z == 1) ? k  : v;

    const int lane   = threadIdx.x & 31;
    const int tileId = blockIdx.x * (blockDim.x >> 5) + (threadIdx.x >> 5);
    const int tm = (tileId >> 5) << 4;   // row tile (o)
    const int tn = (tileId & 31) << 4;   // col tile (t)

    const int laneM = lane & 15;
    const int laneK = (lane >> 4) << 1;  // 0 or 2

    const float* Arow = W + (tm + laneM) * DD + laneK;  // A[m, k..k+1]
    const float* Brow = x + (tn + laneM) * DD + laneK;  // B[k, n] = x[n, k]

    v8f c = {};
#pragma unroll 8
    for (int k0 = 0; k0 < DD; k0 += 4) {
        v2f a = *(const v2f*)(Arow + k0);
        v2f b = *(const v2f*)(Brow + k0);
        c = __builtin_amdgcn_wmma_f32_16x16x4_f32(
                /*neg_a=*/false, a, /*neg_b=*/false, b,
                /*c_mod=*/(short)0, c, /*reuse_a=*/false, /*reuse_b=*/false);
    }

    const int rowOff = (lane >> 4) << 3;  // +8 rows for upper half-wave
    const int col = tn + laneM;
#pragma unroll
    for (int i = 0; i < 8; ++i)
        dst[(tm + rowOff + i) * DD + col] = c[i];
}

// ---------------- Phase 2: RoPE with heads-axis rotate_half ----------------
// o = h*64 + c; partner head is h+4 (h<4, sign -1) or h-4 (h>=4, sign +1).
// angle(c,t) = t * 10000^(-2*(c&31)/64)
__global__ void rope_heads(const float* __restrict__ qin,
                           const float* __restrict__ kin,
                           float* __restrict__ qout,
                           float* __restrict__ kout) {
    int idx = blockIdx.x * blockDim.x + threadIdx.x;   // idx = o*T + t
    int t = idx & (TT - 1);
    int o = idx >> 9;
    int c = o & 63;
    int h = o >> 6;

    float invf = __powf(10000.0f, -(float)((c & 31) << 1) * (1.0f / 64.0f));
    float ang  = (float)t * invf;
    float cs = __cosf(ang);
    float sn = __sinf(ang);

    int poff  = (h < 4) ? (256 * TT) : (-256 * TT);
    float sgn = (h < 4) ? -1.0f : 1.0f;

    qout[idx] = qin[idx] * cs + sgn * qin[idx + poff] * sn;
    kout[idx] = kin[idx] * cs + sgn * kin[idx + poff] * sn;
}

// ---------------- Phase 3: per-channel causal softmax rows ----------------
// One wave handles one (channel o, query i) row: softmax_j<=i( q[o,i]*k[o,j]/8 )
// then weighted sum with v[o,j]. Two-pass (max, then exp-sum) for NaN safety.
__global__ void attn_rows(const float* __restrict__ q,
                          const float* __restrict__ k,
                          const float* __restrict__ v,
                          float* __restrict__ o) {
    const int lane = threadIdx.x & 31;
    const int row  = blockIdx.x * (blockDim.x >> 5) + (threadIdx.x >> 5);
    const int ch = row >> 9;        // channel o in [0,512)
    const int i  = row & (TT - 1);  // query position

    const float* kr = k + ch * TT;
    const float* vr = v + ch * TT;
    const float a = q[ch * TT + i] * 0.125f;   // 1/sqrt(64)

    // pass 1: row max over j<=i
    float m = -3.4e38f;
    for (int j = lane; j <= i; j += 32)
        m = fmaxf(m, a * kr[j]);
#pragma unroll
    for (int off = 16; off > 0; off >>= 1)
        m = fmaxf(m, __shfl_xor(m, off, 32));

    // pass 2: exp-sum and weighted v accumulation
    float s = 0.0f, acc = 0.0f;
    for (int j = lane; j <= i; j += 32) {
        float e = __expf(a * kr[j] - m);
        s   += e;
        acc += e * vr[j];
    }
#pragma unroll
    for (int off = 16; off > 0; off >>= 1) {
        s   += __shfl_xor(s, off, 32);
        acc += __shfl_xor(acc, off, 32);
    }

    if (lane == 0)
        o[ch * TT + i] = acc / s;
}

// ---------------- Phase 4: final = Wo x attout -> d_out[t, o] ----------------
// C[o,t] = sum_d Wo[o,d] * ao[d,t]; stored transposed as out[t*512 + o].
__global__ void out_gemm_f32(const float* __restrict__ ao,
                             const float* __restrict__ Wo,
                             float* __restrict__ out) {
    const int lane   = threadIdx.x & 31;
    const int tileId = blockIdx.x * (blockDim.x >> 5) + (threadIdx.x >> 5);
    const int tm = (tileId >> 5) << 4;   // o tile
    const int tn = (tileId & 31) << 4;   // t tile

    const int laneM = lane & 15;
    const int laneK = (lane >> 4) << 1;

    const float* Arow = Wo + (tm + laneM) * DD + laneK;         // Wo[m, k..k+1]
    const float* Bcol = ao + laneK * TT + (tn + laneM);         // ao[k, n]

    v8f c = {};
#pragma unroll 8
    for (int k0 = 0; k0 < DD; k0 += 4) {
        v2f a = *(const v2f*)(Arow + k0);
        v2f b;
        b.x = Bcol[k0 * TT];
        b.y = Bcol[k0 * TT + TT];
        c = __builtin_amdgcn_wmma_f32_16x16x4_f32(
                false, a, false, b, (short)0, c, false, false);
    }

    const int rowOff = (lane >> 4) << 3;
    const int tcol = tn + laneM;
#pragma unroll
    for (int i = 0; i < 8; ++i)
        out[tcol * DD + (tm + rowOff + i)] = c[i];
}

extern "C" void kernel_launch(void* const* d_in, const int* in_sizes, int n_in,
                              void* d_out, int out_size, void* d_ws, size_t ws_size,
                              hipStream_t stream) {
    const float* x  = (const float*)d_in[0];
    const float* Wq = (const float*)d_in[1];
    const float* Wk = (const float*)d_in[2];
    const float* Wv = (const float*)d_in[3];
    const float* Wo = (const float*)d_in[4];
    float* out = (float*)d_out;

    float* ws = (float*)d_ws;
    const size_t S = (size_t)DD * TT;     // 262144 floats = 1 MB
    float* qraw = ws + 0 * S;
    float* kraw = ws + 1 * S;
    float* vbuf = ws + 2 * S;
    float* qrot = ws + 3 * S;
    float* krot = ws + 4 * S;
    float* atto = qraw;                   // qraw dead after rope; reuse (5 MB total)

    // 1024 16x16 tiles per GEMM, 4 waves (tiles) per 128-thread block.
    qkv_gemm_f32<<<dim3(256, 1, 3), 128, 0, stream>>>(x, Wq, Wk, Wv, qraw, kraw, vbuf);

    // 512*512 elements, elementwise.
    rope_heads<<<(DD * TT) / 256, 256, 0, stream>>>(qraw, kraw, qrot, krot);

    // 512*512 softmax rows, 8 waves per 256-thread block.
    attn_rows<<<(DD * TT) / 8, 256, 0, stream>>>(qrot, krot, vbuf, atto);

    // final projection, transposed store into [T, D] output.
    out_gemm_f32<<<256, 128, 0, stream>>>(atto, Wo, out);
}